// DeformTransLayer_71184787964531
// MI455X (gfx1250) — compile-verified
//
#include <hip/hip_runtime.h>
#include <math.h>

// ---------------------------------------------------------------------------
// CDNA5 (gfx1250) deformable transformer layer.
// Matrix math uses V_WMMA_F32_16X16X4_F32 (wave32 WMMA, fp32 in/out),
// software-pipelined so fragment loads for step k+1 overlap WMMAs of step k.
// ---------------------------------------------------------------------------

typedef float v2f __attribute__((ext_vector_type(2)));
typedef float v8f __attribute__((ext_vector_type(8)));

#define WMMA_F32(a, b, c) \
    __builtin_amdgcn_wmma_f32_16x16x4_f32(false, (a), false, (b), (short)0, (c), false, false)

// ---------------------------------------------------------------------------
// Generic fp32 WMMA GEMM:  C[M,N] = act( A[M,K] * B[K,N] + bias [+ Cprev] )
// A row-major (lda), B row-major (ldb), C row-major (ldc).
// One wave computes a 32(M) x 64(N) tile (2 row-frags x 4 col-frags,
// 8 accumulators); 4 waves / block -> 128 rows x 64 cols per block.
// Fragment layouts per CDNA5 ISA 7.12.2:
//   A 16x4 : lane l -> m = l%16, k = 2*(l/16) + vgpr   (per-lane float2 at k0+2*half)
//   B 4x16 : lane l -> n = l%16, k = 2*(l/16) + vgpr
//   C 16x16: lane l, vgpr r -> m = r + 8*(l/16), n = l%16
// Requires: K % 4 == 0, N % 64 == 0 (true for every call below).
// ---------------------------------------------------------------------------
__global__ __launch_bounds__(128)
void gemm_wmma_f32(const float* __restrict__ A, int lda,
                   const float* __restrict__ B, int ldb,
                   const float* __restrict__ bias,
                   float* __restrict__ C, int ldc,
                   int M, int N, int K, int do_relu, int accum)
{
    const int lane  = threadIdx.x & 31;
    const int wave  = threadIdx.x >> 5;
    const int idx16 = lane & 15;
    const int half  = lane >> 4;          // 0 or 1

    const int rowTile = blockIdx.x * 4 + wave;
    const int row0    = rowTile * 32;
    if (row0 >= M) return;                // uniform per wave -> EXEC stays all-ones
    const int col0    = blockIdx.y * 64;

    int ar0 = row0 + idx16;       if (ar0 >= M) ar0 = M - 1;
    int ar1 = row0 + 16 + idx16;  if (ar1 >= M) ar1 = M - 1;
    const float* __restrict__ A0 = A + (size_t)ar0 * (size_t)lda;
    const float* __restrict__ A1 = A + (size_t)ar1 * (size_t)lda;

    v8f acc[2][4] = {};

    // fragment loader: A row-pair (float2) + 4 B column-tile fragments
    auto LOAD = [&](int k0, v2f& a0, v2f& a1, v2f* b) {
        const int kb = k0 + 2 * half;     // even -> 8B-aligned float2 load
        a0 = *(const v2f*)(A0 + kb);
        a1 = *(const v2f*)(A1 + kb);
        const float* __restrict__ Bp0 = B + (size_t)kb * (size_t)ldb + col0 + idx16;
        const float* __restrict__ Bp1 = Bp0 + ldb;
#pragma unroll
        for (int t = 0; t < 4; ++t) { b[t].x = Bp0[t * 16]; b[t].y = Bp1[t * 16]; }
    };

    v2f a0c, a1c, bc[4];
    LOAD(0, a0c, a1c, bc);

    int k0 = 0;
    for (; k0 + 4 < K; k0 += 4) {
        v2f a0n, a1n, bn[4];
        LOAD(k0 + 4, a0n, a1n, bn);       // issue next-step loads first
#pragma unroll
        for (int t = 0; t < 4; ++t) {     // then 8 back-to-back WMMAs on current frags
            acc[0][t] = WMMA_F32(a0c, bc[t], acc[0][t]);
            acc[1][t] = WMMA_F32(a1c, bc[t], acc[1][t]);
        }
        a0c = a0n; a1c = a1n;
#pragma unroll
        for (int t = 0; t < 4; ++t) bc[t] = bn[t];
    }
#pragma unroll
    for (int t = 0; t < 4; ++t) {         // drain last step
        acc[0][t] = WMMA_F32(a0c, bc[t], acc[0][t]);
        acc[1][t] = WMMA_F32(a1c, bc[t], acc[1][t]);
    }

#pragma unroll
    for (int h2 = 0; h2 < 2; ++h2) {
#pragma unroll
        for (int t = 0; t < 4; ++t) {
            const int col = col0 + t * 16 + idx16;
            const float bv = bias ? bias[col] : 0.0f;
#pragma unroll
            for (int r = 0; r < 8; ++r) {
                const int row = row0 + h2 * 16 + r + 8 * half;
                if (row < M) {
                    const size_t o = (size_t)row * (size_t)ldc + col;
                    float v = acc[h2][t][r] + bv;
                    if (accum)   v += C[o];
                    if (do_relu) v = fmaxf(v, 0.0f);
                    C[o] = v;
                }
            }
        }
    }
}

// ---------------------------------------------------------------------------
// Sampling locations + softmax attention weights.
// One thread per (n,q,m): softmax over L*P=32 logits, and 32 (x,y) locations.
// Writes directly into the loc / attn slices of d_out.
// ---------------------------------------------------------------------------
__global__ __launch_bounds__(256)
void loc_attn_kernel(const float* __restrict__ off,    // [NLq, 256]
                     const float* __restrict__ logit,  // [NLq, 128]
                     const float* __restrict__ refp,   // [NLq, 4, 2]
                     const int*   __restrict__ ss,     // [4,2] (H,W)
                     float* __restrict__ loc_out,      // [NLq, 4, 4, 8, 2]
                     float* __restrict__ attn_out,     // [NLq, 4, 4, 8]
                     int NLq)
{
    const int t = blockIdx.x * blockDim.x + threadIdx.x;
    if (t >= NLq * 4) return;
    const int m  = t & 3;
    const int nq = t >> 2;

    // softmax over 32
    const float* __restrict__ lg = logit + (size_t)nq * 128 + m * 32;
    float v[32];
    float mx = -3.0e38f;
#pragma unroll
    for (int i = 0; i < 32; ++i) { v[i] = lg[i]; mx = fmaxf(mx, v[i]); }
    float sum = 0.0f;
#pragma unroll
    for (int i = 0; i < 32; ++i) { v[i] = __expf(v[i] - mx); sum += v[i]; }
    const float inv = 1.0f / sum;
    float* __restrict__ ao = attn_out + (size_t)t * 32;
#pragma unroll
    for (int i = 0; i < 32; ++i) ao[i] = v[i] * inv;

    // locations: loc = ref + off / (W,H)
    const float* __restrict__ ob = off + (size_t)nq * 256 + m * 64;
    float* __restrict__ lo = loc_out + (size_t)t * 64;
#pragma unroll
    for (int l = 0; l < 4; ++l) {
        const float invW = 1.0f / (float)ss[2 * l + 1];
        const float invH = 1.0f / (float)ss[2 * l + 0];
        const float rx = refp[((size_t)nq * 4 + l) * 2 + 0];
        const float ry = refp[((size_t)nq * 4 + l) * 2 + 1];
#pragma unroll
        for (int p = 0; p < 8; ++p) {
            lo[l * 16 + p * 2 + 0] = rx + ob[l * 16 + p * 2 + 0] * invW;
            lo[l * 16 + p * 2 + 1] = ry + ob[l * 16 + p * 2 + 1] * invH;
        }
    }
}

// ---------------------------------------------------------------------------
// Multi-scale deformable bilinear sampling + attention-weighted sum.
// One thread per output element (n,q,m,d); 32 samples * 4 corners each.
// grid_sample semantics: align_corners=False, padding_mode='zeros'.
// ---------------------------------------------------------------------------
__global__ __launch_bounds__(256)
void msdeform_sample(const float* __restrict__ value, // [N*Len, 256] head-interleaved
                     const float* __restrict__ loc,   // [NLq*4, 64]
                     const float* __restrict__ attn,  // [NLq*4, 32]
                     const int*   __restrict__ ss,    // [4,2]
                     const int*   __restrict__ lsi,   // [4]
                     float* __restrict__ outbuf,      // [NLq, 256]
                     int Nbatch, int Len)
{
    const size_t t = (size_t)blockIdx.x * blockDim.x + threadIdx.x;
    const size_t total = (size_t)Nbatch * Len * 256;
    if (t >= total) return;
    const int    d  = (int)(t & 63);
    const int    m  = (int)((t >> 6) & 3);
    const size_t nq = t >> 8;                 // n*Len + q
    const int    n  = (int)(nq / (size_t)Len);

    const float* __restrict__ lo = loc  + (nq * 4 + m) * 64;
    const float* __restrict__ at = attn + (nq * 4 + m) * 32;
    const float* __restrict__ vb = value + (size_t)n * Len * 256 + m * 64 + d;

    float acc = 0.0f;
#pragma unroll
    for (int l = 0; l < 4; ++l) {
        const int H = ss[2 * l], W = ss[2 * l + 1];
        const int start = lsi[l];
#pragma unroll
        for (int p = 0; p < 8; ++p) {
            const float x = lo[l * 16 + p * 2 + 0] * (float)W - 0.5f;
            const float y = lo[l * 16 + p * 2 + 1] * (float)H - 0.5f;
            const float xf = floorf(x), yf = floorf(y);
            const float wx = x - xf, wy = y - yf;
            const int x0 = (int)xf, y0 = (int)yf;
            float s = 0.0f;
            if (y0 >= 0 && y0 < H) {
                const float* r0 = vb + (size_t)(start + y0 * W) * 256;
                if (x0 >= 0 && x0 < W)         s += (1.0f - wx) * (1.0f - wy) * r0[(size_t)x0 * 256];
                if (x0 + 1 >= 0 && x0 + 1 < W) s += wx * (1.0f - wy) * r0[(size_t)(x0 + 1) * 256];
            }
            if (y0 + 1 >= 0 && y0 + 1 < H) {
                const float* r1 = vb + (size_t)(start + (y0 + 1) * W) * 256;
                if (x0 >= 0 && x0 < W)         s += (1.0f - wx) * wy * r1[(size_t)x0 * 256];
                if (x0 + 1 >= 0 && x0 + 1 < W) s += wx * wy * r1[(size_t)(x0 + 1) * 256];
            }
            acc += at[l * 8 + p] * s;
        }
    }
    outbuf[nq * 256 + m * 64 + d] = acc;
}

// ---------------------------------------------------------------------------
// out = LayerNorm(a + b) * g + be   over D_MODEL=256, one block per row.
// ---------------------------------------------------------------------------
__global__ __launch_bounds__(256)
void add_layernorm(const float* __restrict__ a, const float* __restrict__ b,
                   const float* __restrict__ g, const float* __restrict__ be,
                   float* __restrict__ out)
{
    __shared__ float red[256];
    const int row = blockIdx.x;
    const int tid = threadIdx.x;
    const size_t o = (size_t)row * 256 + tid;
    const float s = a[o] + b[o];
    red[tid] = s;
    __syncthreads();
#pragma unroll
    for (int off = 128; off > 0; off >>= 1) {
        if (tid < off) red[tid] += red[tid + off];
        __syncthreads();
    }
    const float mean = red[0] * (1.0f / 256.0f);
    __syncthreads();
    const float dd = s - mean;
    red[tid] = dd * dd;
    __syncthreads();
#pragma unroll
    for (int off = 128; off > 0; off >>= 1) {
        if (tid < off) red[tid] += red[tid + off];
        __syncthreads();
    }
    const float var = red[0] * (1.0f / 256.0f);
    out[o] = dd * rsqrtf(var + 1e-5f) * g[tid] + be[tid];
}

// ---------------------------------------------------------------------------
// Host-side orchestration (graph-capture safe: only kernel launches on stream)
// ---------------------------------------------------------------------------
static inline void launch_gemm(const float* A, int lda, const float* B, int ldb,
                               const float* bias, float* C, int ldc,
                               int M, int N, int K, int relu, int accum,
                               hipStream_t stream)
{
    dim3 grid((unsigned)(((M + 31) / 32 + 3) / 4), (unsigned)(N / 64));
    gemm_wmma_f32<<<grid, 128, 0, stream>>>(A, lda, B, ldb, bias, C, ldc,
                                            M, N, K, relu, accum);
}

extern "C" void kernel_launch(void* const* d_in, const int* in_sizes, int n_in,
                              void* d_out, int out_size, void* d_ws, size_t ws_size,
                              hipStream_t stream)
{
    const float* src    = (const float*)d_in[0];   // (N, Len, 256)
    const float* refp   = (const float*)d_in[1];   // (N, Len, 4, 2)
    const float* keyf   = (const float*)d_in[2];   // (N, Len, 256)
    const int*   ss     = (const int*)  d_in[3];   // (4,2)
    const int*   lsi    = (const int*)  d_in[4];   // (4,)
    const float* W_off  = (const float*)d_in[5];   // (256,256)
    const float* b_off  = (const float*)d_in[6];
    const float* W_attn = (const float*)d_in[7];   // (256,128)
    const float* b_attn = (const float*)d_in[8];
    const float* W_val  = (const float*)d_in[9];   // (256,256)
    const float* b_val  = (const float*)d_in[10];
    const float* W_out  = (const float*)d_in[11];  // (256,256)
    const float* b_out  = (const float*)d_in[12];
    const float* W1     = (const float*)d_in[13];  // (256,1024)
    const float* b1     = (const float*)d_in[14];
    const float* W2     = (const float*)d_in[15];  // (1024,256)
    const float* b2     = (const float*)d_in[16];
    const float* g1     = (const float*)d_in[17];
    const float* be1    = (const float*)d_in[18];
    const float* g2     = (const float*)d_in[19];
    const float* be2    = (const float*)d_in[20];

    const int NLq = in_sizes[0] / 256;   // N*Len = 40394
    const int Nb  = 2;                   // BATCH (static in reference)
    const int Len = NLq / Nb;

    // workspace: three [NLq,256] fp32 buffers (~41.4 MB each)
    const size_t BUF = (size_t)NLq * 256;
    float* buf0 = (float*)d_ws;          // value  -> x1
    float* buf1 = buf0 + BUF;            // off    -> msout -> ffn_y
    float* buf2 = buf1 + BUF;            // logits -> src2  -> ffn_h

    // d_out slices: [ x (NLq*256) | loc (NLq*256) | attn (NLq*128) ]
    float* out_x    = (float*)d_out;
    float* out_loc  = out_x   + BUF;
    float* out_attn = out_loc + BUF;

    // 1) value = key_feat @ W_val + b_val
    launch_gemm(keyf, 256, W_val, 256, b_val, buf0, 256, NLq, 256, 256, 0, 0, stream);
    // 2) off = src @ W_off + b_off
    launch_gemm(src, 256, W_off, 256, b_off, buf1, 256, NLq, 256, 256, 0, 0, stream);
    // 3) attn logits = src @ W_attn + b_attn
    launch_gemm(src, 256, W_attn, 128, b_attn, buf2, 128, NLq, 128, 256, 0, 0, stream);

    // 4) locations + softmax -> d_out loc/attn
    {
        const int tot = NLq * 4;
        loc_attn_kernel<<<(tot + 255) / 256, 256, 0, stream>>>(
            buf1, buf2, refp, ss, out_loc, out_attn, NLq);
    }

    // 5) deformable sampling -> buf1 (msout, head-interleaved [NLq,256])
    {
        const size_t tot = (size_t)NLq * 256;
        msdeform_sample<<<(unsigned)((tot + 255) / 256), 256, 0, stream>>>(
            buf0, out_loc, out_attn, ss, lsi, buf1, Nb, Len);
    }

    // 6) src2 = msout @ W_out + b_out -> buf2
    launch_gemm(buf1, 256, W_out, 256, b_out, buf2, 256, NLq, 256, 256, 0, 0, stream);

    // 7) x1 = LN(src + src2) -> buf0
    add_layernorm<<<NLq, 256, 0, stream>>>(src, buf2, g1, be1, buf0);

    // 8) FFN, chunked over D_FFN=1024 in 4 chunks of 256 (accumulate into buf1)
    for (int c = 0; c < 4; ++c) {
        // h = relu(x1 @ W1[:, c*256:(c+1)*256] + b1[chunk]) -> buf2
        launch_gemm(buf0, 256, W1 + c * 256, 1024, b1 + c * 256,
                    buf2, 256, NLq, 256, 256, 1, 0, stream);
        // y (+)= h @ W2[c*256:(c+1)*256, :]  (+ b2 on first chunk)
        launch_gemm(buf2, 256, W2 + (size_t)c * 256 * 256, 256,
                    (c == 0) ? b2 : nullptr,
                    buf1, 256, NLq, 256, 256, 0, (c == 0) ? 0 : 1, stream);
    }

    // 9) x = LN(x1 + ffn) -> d_out
    add_layernorm<<<NLq, 256, 0, stream>>>(buf0, buf1, g2, be2, out_x);
}